// BinarySCLoss_67207648248312
// MI455X (gfx1250) — compile-verified
//
#include <hip/hip_runtime.h>
#include <cstdint>

#define TILE_W   32
#define TILE_H   8
#define HALO     2
#define PW       (TILE_W + 2*HALO)   // 36
#define PH       (TILE_H + 2*HALO)   // 12
#define PN       (PW * PH)           // 432
#define NTHREADS 256
#define IMG_H    512
#define IMG_W    512

#if __has_builtin(__builtin_amdgcn_tensor_load_to_lds)
#define HAVE_TDM 1
#else
#define HAVE_TDM 0
#endif

typedef unsigned int u32x4 __attribute__((ext_vector_type(4)));
typedef int          i32x4 __attribute__((ext_vector_type(4)));
typedef int          i32x8 __attribute__((ext_vector_type(8)));

// stable softplus: log(1+exp(x)) = max(x,0) + log(1+exp(-|x|)), fast hw exp/log
__device__ __forceinline__ float fast_softplus(float x) {
    float e = __expf(-fabsf(x));
    return fmaxf(x, 0.0f) + __logf(1.0f + e);
}

#if HAVE_TDM
// Issue one TDM descriptor: 2D tile (tileW x tileH elems, 4B each) from global
// (row stride strideElems) into LDS at lds_off. D# packing per CDNA5 ISA 8.3/8.4.
__device__ __forceinline__ void tdm_load_tile_f32(
    uint32_t lds_off, const float* gtile, int tensorW, int tensorH,
    int tileW, int tileH, int strideElems)
{
    const uint64_t ga = (uint64_t)(uintptr_t)gtile;
    u32x4 g0;
    g0[0] = 1u;                                   // count=1, user mode, no gather
    g0[1] = lds_off;                              // lds_addr [63:32]
    g0[2] = (uint32_t)(ga & 0xffffffffu);         // global_addr[31:0]
    g0[3] = (uint32_t)((ga >> 32) & 0x1ffffffu)   // global_addr[56:32]
          | (2u << 30);                           // type=2 ("image")
    i32x8 g1;
    g1[0] = (int)(2u << 16);                      // data_size=2 (4B); no multicast/pad/iter
    g1[1] = (int)(((uint32_t)tensorW & 0xffffu) << 16);            // tensor_dim0[15:0]
    g1[2] = (int)((((uint32_t)tensorW >> 16) & 0xffffu)
          | (((uint32_t)tensorH & 0xffffu) << 16));                // dim0[31:16], dim1[15:0]
    g1[3] = (int)((((uint32_t)tensorH >> 16) & 0xffffu)
          | (((uint32_t)tileW & 0xffffu) << 16));                  // dim1[31:16], tile_dim0
    g1[4] = (int)((uint32_t)tileH & 0xffffu);                      // tile_dim1; tile_dim2=0
    g1[5] = strideElems;                                           // tensor_dim0_stride[31:0]
    g1[6] = 0;                                                     // stride[47:32]; dim1_stride lo
    g1[7] = 0;                                                     // dim1_stride hi (unused, 2D)
    const i32x4 z4 = {0, 0, 0, 0};                                 // groups 2/3: unused (2D)
    const i32x8 z8 = {0, 0, 0, 0, 0, 0, 0, 0};
    __builtin_amdgcn_tensor_load_to_lds(g0, g1, z4, z4, z8, 0);
}
#endif

__global__ __launch_bounds__(NTHREADS) void bsc_loss_main(
    const float* __restrict__ pred, const float* __restrict__ target,
    float* __restrict__ partial, int H, int W)
{
    __shared__ float sP[PN];   // pred tile (with halo)
    __shared__ float sT[PN];   // target tile
    __shared__ float sS[PN];   // sigmoid(pred) tile
    __shared__ float sRed[NTHREADS / 32];

    const int tx  = (int)(threadIdx.x % TILE_W);
    const int ty  = (int)(threadIdx.x / TILE_W);
    const int gx0 = (int)blockIdx.x * TILE_W;
    const int gy0 = (int)blockIdx.y * TILE_H;
    const size_t imgBase = (size_t)blockIdx.z * (size_t)H * (size_t)W;
    const float* __restrict__ pbase = pred + imgBase;
    const float* __restrict__ tbase = target + imgBase;

    // Uniform (scalar-branch) predicate: halo tile fully inside the image?
    const bool interior =
        (blockIdx.x >= 1) && (blockIdx.x <= gridDim.x - 2) &&
        (blockIdx.y >= 1) && (blockIdx.y <= gridDim.y - 2);

#if HAVE_TDM
    if (interior) {
        // --- CDNA5 Tensor Data Mover: one 36x12 strided tile per array ---
        // TDM ignores EXEC, so gate on a *scalar* wave-id condition to issue
        // exactly once (wave 0); its s_wait_tensorcnt + the workgroup barrier
        // make the LDS writes visible to all 8 waves.
        if (__builtin_amdgcn_readfirstlane((int)threadIdx.x) < 32) {
            const int tox = gx0 - HALO, toy = gy0 - HALO;
            tdm_load_tile_f32((uint32_t)(uintptr_t)&sP[0],
                              pbase + (size_t)toy * W + tox, W, H, PW, PH, W);
            tdm_load_tile_f32((uint32_t)(uintptr_t)&sT[0],
                              tbase + (size_t)toy * W + tox, W, H, PW, PH, W);
            __builtin_amdgcn_s_wait_tensorcnt(0);
        }
    } else
#endif
    {
        // --- border blocks: per-lane async global->LDS with clamped coords ---
        for (int l = (int)threadIdx.x; l < PN; l += NTHREADS) {
            int lx = l % PW, ly = l / PW;
            int gx = min(max(gx0 - HALO + lx, 0), W - 1);
            int gy = min(max(gy0 - HALO + ly, 0), H - 1);
            uint32_t goff = (uint32_t)((gy * W + gx) * (int)sizeof(float));
            uint32_t ldsP = (uint32_t)(uintptr_t)(&sP[l]);
            uint32_t ldsT = (uint32_t)(uintptr_t)(&sT[l]);
            asm volatile("global_load_async_to_lds_b32 %0, %1, %2"
                         :: "v"(ldsP), "v"(goff), "s"(pbase) : "memory");
            asm volatile("global_load_async_to_lds_b32 %0, %1, %2"
                         :: "v"(ldsT), "v"(goff), "s"(tbase) : "memory");
        }
        asm volatile("s_wait_asynccnt 0" ::: "memory");
    }
    __syncthreads();

    // --- precompute sigmoid(pred) once per tile element ---
    for (int l = (int)threadIdx.x; l < PN; l += NTHREADS) {
        sS[l] = __frcp_rn(1.0f + __expf(-sP[l]));
    }
    __syncthreads();

    // --- per-pixel 5x5 evaluation; 3x3 window shares denominators with 5x5 ---
    const int cx = tx + HALO, cy = ty + HALO;
    const int gx = gx0 + tx,  gy = gy0 + ty;
    const int cidx = cy * PW + cx;
    const float pc  = sP[cidx];
    const float tc  = sT[cidx];
    const float sc  = sS[cidx];
    const float bce = fast_softplus(pc) - pc * tc;

    float acc = 0.0f;
#pragma unroll
    for (int dy = -2; dy <= 2; ++dy) {
#pragma unroll
        for (int dx = -2; dx <= 2; ++dx) {
            // weight: i=1 term (1/9) only for inner 3x3; i=2 term 0.5/25 for all 25
            const float w = (((dy >= -1) && (dy <= 1) && (dx >= -1) && (dx <= 1))
                                 ? (1.0f / 9.0f) : 0.0f) + 0.02f;
            const int   idx = (cy + dy) * PW + (cx + dx);
            const float pn  = sP[idx];
            const float tn  = sT[idx];
            const float sn  = sS[idx];
            const float j   = pc * pn;
            const float mut = fast_softplus(j) - j * (tc * tn);
            const float pw  = __expf(-sc * sn);
            const float loss = bce * __frcp_rn(mut + pw + 1e-6f);
            const bool valid = ((unsigned)(gx + dx) < (unsigned)W) &&
                               ((unsigned)(gy + dy) < (unsigned)H);
            acc += w * (valid ? loss : 1.0f);
        }
    }

    // --- deterministic block reduction (wave32 butterfly + LDS across 8 waves) ---
#pragma unroll
    for (int off = 16; off > 0; off >>= 1)
        acc += __shfl_xor(acc, off, 32);
    const int lane = (int)(threadIdx.x & 31u);
    const int wid  = (int)(threadIdx.x >> 5u);
    if (lane == 0) sRed[wid] = acc;
    __syncthreads();
    if (threadIdx.x == 0) {
        float s = 0.0f;
#pragma unroll
        for (int i = 0; i < NTHREADS / 32; ++i) s += sRed[i];
        partial[blockIdx.x + gridDim.x * (blockIdx.y + gridDim.y * blockIdx.z)] = s;
    }
}

__global__ __launch_bounds__(NTHREADS) void bsc_loss_reduce(
    const float* __restrict__ partial, float* __restrict__ out, int n, float scale)
{
    __shared__ float sRed[NTHREADS / 32];
    float acc = 0.0f;
    for (int i = (int)threadIdx.x; i < n; i += NTHREADS) acc += partial[i];
#pragma unroll
    for (int off = 16; off > 0; off >>= 1)
        acc += __shfl_xor(acc, off, 32);
    const int lane = (int)(threadIdx.x & 31u);
    const int wid  = (int)(threadIdx.x >> 5u);
    if (lane == 0) sRed[wid] = acc;
    __syncthreads();
    if (threadIdx.x == 0) {
        float s = 0.0f;
#pragma unroll
        for (int i = 0; i < NTHREADS / 32; ++i) s += sRed[i];
        out[0] = s * scale;
    }
}

extern "C" void kernel_launch(void* const* d_in, const int* in_sizes, int n_in,
                              void* d_out, int out_size, void* d_ws, size_t ws_size,
                              hipStream_t stream) {
    const float* pred   = (const float*)d_in[0];
    const float* target = (const float*)d_in[1];
    const int H = IMG_H, W = IMG_W;
    const int B = in_sizes[0] / (H * W);   // 8 for the reference shapes

    dim3 block(NTHREADS, 1, 1);
    dim3 grid(W / TILE_W, H / TILE_H, B);  // 16 x 64 x 8 = 8192 blocks
    float* partial = (float*)d_ws;         // needs 8192 * 4 B = 32 KB of scratch

    bsc_loss_main<<<grid, block, 0, stream>>>(pred, target, partial, H, W);

    const int nPart = (int)(grid.x * grid.y * grid.z);
    const float scale = 1.0f / ((float)B * (float)H * (float)W);
    bsc_loss_reduce<<<1, block, 0, stream>>>(partial, (float*)d_out, nPart, scale);
}